// PrototypicalNetworks_76312978915590
// MI455X (gfx1250) — compile-verified
//
#include <hip/hip_runtime.h>
#include <hip/hip_bf16.h>
#include <stdint.h>

// ---------------------------------------------------------------------------
// Prototypical networks, MI455X (gfx1250, wave32, WMMA + async-to-LDS).
//
// inputs:
//   d_in[0] support_images  [65536,1024] f32
//   d_in[1] support_labels  [65536, 256] i32 (one-hot)
//   d_in[2] query_images    [65536,1024] f32
//   d_in[3] W               [1024, 512] f32
//   d_in[4] b               [512]       f32
// output:  softmax scores   [65536, 256] f32
// ---------------------------------------------------------------------------

typedef float v2f __attribute__((ext_vector_type(2)));
typedef float v8f __attribute__((ext_vector_type(8)));
typedef int   v4i __attribute__((ext_vector_type(4)));

#define NS     65536
#define NQ     65536
#define FEAT   1024
#define EMB    512
#define WAYS   256
#define CHUNKS 64
#define RPC    (NS / CHUNKS)

// ---- f32 WMMA (16x16x4, D = A*B + C) --------------------------------------
__device__ __forceinline__ v8f wmma_f32(v2f a, v2f b, v8f c) {
    return __builtin_amdgcn_wmma_f32_16x16x4_f32(
        false, a, false, b, (short)0, c, false, false);
}

// ---- async global -> LDS, 16 bytes per lane (ASYNCcnt-tracked) ------------
typedef __attribute__((address_space(1))) v4i gv4i;   // global
typedef __attribute__((address_space(3))) v4i lv4i;   // LDS

__device__ __forceinline__ void async_copy16(const float* g, float* l) {
#if __has_builtin(__builtin_amdgcn_global_load_async_to_lds_b128)
    __builtin_amdgcn_global_load_async_to_lds_b128((gv4i*)g, (lv4i*)l, 0, 0);
#else   // fallback: synchronous copy (pipeline structure remains correct)
    *(float4*)l = *(const float4*)g;
#endif
}
// wait until at most N async ops outstanding (loads complete in order)
#define WAIT_ASYNC(n) asm volatile("s_wait_asynccnt %0" :: "n"(n) : "memory")

// ---------------------------------------------------------------------------
__global__ void k_zero_counts(int* __restrict__ counts) {
    if (threadIdx.x < WAYS) counts[threadIdx.x] = 0;
}

// labels = argmax(one-hot); counts via integer atomics (deterministic)
__global__ void k_labels(const int* __restrict__ onehot,
                         int* __restrict__ labels, int* __restrict__ counts) {
    int row = blockIdx.x * 256 + threadIdx.x;
    const int4* p = (const int4*)(onehot + (size_t)row * WAYS);
    int lab = 0;
    #pragma unroll 8
    for (int j = 0; j < WAYS / 4; ++j) {
        int4 v = p[j];
        if (v.x) lab = 4 * j + 0;
        if (v.y) lab = 4 * j + 1;
        if (v.z) lab = 4 * j + 2;
        if (v.w) lab = 4 * j + 3;
    }
    labels[row] = lab;
    atomicAdd(&counts[lab], 1);
}

// per-chunk class sums: owner-computes (thread t owns feature column t of all
// 256 class accumulators, 256 KB LDS), rows sequential -> deterministic.
__global__ void k_class_partial(const float* __restrict__ S,
                                const int* __restrict__ labels,
                                float* __restrict__ partials) {
    extern __shared__ float acc[];                 // [WAYS][256]
    const int t     = threadIdx.x;
    const int chunk = blockIdx.x;
    const int slice = blockIdx.y;
    for (int i = t; i < WAYS * 256; i += 256) acc[i] = 0.f;
    __syncthreads();
    const int r0 = chunk * RPC;
    const size_t fbase = (size_t)slice * 256 + t;
    for (int r = 0; r < RPC; r += 4) {
        int   l0 = labels[r0 + r + 0];
        int   l1 = labels[r0 + r + 1];
        int   l2 = labels[r0 + r + 2];
        int   l3 = labels[r0 + r + 3];
        float v0 = S[(size_t)(r0 + r + 0) * FEAT + fbase];
        float v1 = S[(size_t)(r0 + r + 1) * FEAT + fbase];
        float v2 = S[(size_t)(r0 + r + 2) * FEAT + fbase];
        float v3 = S[(size_t)(r0 + r + 3) * FEAT + fbase];
        acc[l0 * 256 + t] += v0;
        acc[l1 * 256 + t] += v1;
        acc[l2 * 256 + t] += v2;
        acc[l3 * 256 + t] += v3;
    }
    __syncthreads();
    for (int c = 0; c < WAYS; ++c)
        partials[((size_t)chunk * WAYS + c) * FEAT + fbase] = acc[c * 256 + t];
}

__global__ void k_reduce_mean(const float* __restrict__ partials,
                              const int* __restrict__ counts,
                              float* __restrict__ smean) {
    int idx = blockIdx.x * 256 + threadIdx.x;
    int c = idx >> 10, f = idx & 1023;
    float s = 0.f;
    #pragma unroll 8
    for (int k = 0; k < CHUNKS; ++k)
        s += partials[((size_t)k * WAYS + c) * FEAT + f];
    smean[idx] = s / (float)counts[c];
}

// z_proto = smean @ W + b  [256,512]; one wave per 16x16 tile (WMMA f32)
__global__ void k_proto_gemm(const float* __restrict__ smean,
                             const float* __restrict__ W,
                             const float* __restrict__ bvec,
                             float* __restrict__ zproto) {
    const int lane = threadIdx.x;
    const int nt = blockIdx.x, mt = blockIdx.y;
    const int lm = lane & 15;
    const int ka = (lane >> 4) * 2;
    const int m = mt * 16 + lm;
    const int n = nt * 16 + lm;
    v8f acc = {0.f, 0.f, 0.f, 0.f, 0.f, 0.f, 0.f, 0.f};
    for (int k = 0; k < FEAT; k += 4) {
        v2f a = *(const v2f*)&smean[(size_t)m * FEAT + k + ka];
        v2f b;
        b.x = W[(size_t)(k + ka + 0) * EMB + n];
        b.y = W[(size_t)(k + ka + 1) * EMB + n];
        acc = wmma_f32(a, b, acc);
    }
    const float bv = bvec[n];
    const int rbase = (lane >> 4) * 8;
    #pragma unroll
    for (int r = 0; r < 8; ++r) {
        int row = mt * 16 + rbase + r;
        zproto[(size_t)row * EMB + n] = acc[r] + bv;
    }
}

__global__ void k_p2(const float* __restrict__ zp, float* __restrict__ p2) {
    int c = threadIdx.x;
    float s = 0.f;
    for (int e = 0; e < EMB; ++e) { float v = zp[(size_t)c * EMB + e]; s += v * v; }
    p2[c] = s;
}

// ---------------------------------------------------------------------------
// Fused: z_query block (64x512, WMMA, K=1024) -> LDS -> distances vs 256
// protos (WMMA, K=512) -> -sqrt -> row softmax. Async double-buffered slabs.
//
// LDS layout (float offsets):
#define ZQ_ST    514
#define QP_ST    36                     // q/proto slab row stride (144B, 16B-aligned)
#define W_ST     520                    // W slab row stride (2080B, 16B-aligned)
#define SB_ST    257
#define QSL_SZ   (64 * QP_ST)           // 2304
#define WSL_SZ   (32 * W_ST)            // 16640
#define PSL_SZ   (256 * QP_ST)          // 9216
#define OFF_ZQ   0
#define OFF_QSL  32896                  // 64*514
#define OFF_WSL  (OFF_QSL + 2 * QSL_SZ)         // 37504
#define OFF_PSL  OFF_QSL                // stage-2 reuse
#define OFF_SB   (OFF_PSL + 2 * PSL_SZ)         // 51328
#define OFF_RED  (OFF_SB + 64 * SB_ST)          // 67776
#define OFF_RM   (OFF_RED + 256)
#define OFF_RS   (OFF_RM + 64)
#define OFF_Q2   (OFF_RS + 64)
#define BC_LDS_F (OFF_WSL + 2 * WSL_SZ)         // 70784
#define BC_LDS_B (BC_LDS_F * 4)                 // 283136 B  (< 320 KB)

__device__ __forceinline__ void s1_issue(const float* __restrict__ Q,
                                         const float* __restrict__ W,
                                         float* qbuf, float* wbuf,
                                         int q0, int k0, int t) {
    #pragma unroll
    for (int i = 0; i < 2; ++i) {                  // Q slab [64][32]
        int c = t + 256 * i;
        int m = c >> 3, kc = c & 7;
        async_copy16(&Q[(size_t)(q0 + m) * FEAT + k0 + kc * 4],
                     &qbuf[m * QP_ST + kc * 4]);
    }
    #pragma unroll
    for (int i = 0; i < 16; ++i) {                 // W slab [32][512], K-major
        int c = t + 256 * i;
        int kk = c >> 7, nc = c & 127;
        async_copy16(&W[(size_t)(k0 + kk) * EMB + nc * 4],
                     &wbuf[kk * W_ST + nc * 4]);
    }
}

__device__ __forceinline__ void s2_issue(const float* __restrict__ P,
                                         float* pbuf, int k0, int t) {
    #pragma unroll
    for (int i = 0; i < 8; ++i) {                  // proto slab [256][32]
        int c = t + 256 * i;
        int n = c >> 3, kc = c & 7;
        async_copy16(&P[(size_t)n * EMB + k0 + kc * 4],
                     &pbuf[n * QP_ST + kc * 4]);
    }
}

__global__ void k_query_fused(const float* __restrict__ Q,
                              const float* __restrict__ W,
                              const float* __restrict__ bvec,
                              const float* __restrict__ zproto,
                              const float* __restrict__ p2,
                              float* __restrict__ out) {
    extern __shared__ float lds[];
    float* zq  = lds + OFF_ZQ;
    float* qsl = lds + OFF_QSL;
    float* wsl = lds + OFF_WSL;
    float* psl = lds + OFF_PSL;
    float* sb  = lds + OFF_SB;
    float* red = lds + OFF_RED;
    float* rwm = lds + OFF_RM;
    float* rws = lds + OFF_RS;
    float* q2l = lds + OFF_Q2;

    const int t     = threadIdx.x;
    const int lane  = t & 31;
    const int wv    = t >> 5;
    const int lm    = lane & 15;
    const int ka    = (lane >> 4) * 2;
    const int rbase = (lane >> 4) * 8;
    const int q0    = blockIdx.x * 64;

    const v8f vzero = {0.f, 0.f, 0.f, 0.f, 0.f, 0.f, 0.f, 0.f};
    v8f acc[16];

    // ---------------- stage 1: zq = Q[q0:q0+64] @ W + b --------------------
    #pragma unroll
    for (int j = 0; j < 16; ++j) acc[j] = vzero;
    const int mt1 = wv & 3;
    const int ng1 = (wv >> 2) * 16;

    s1_issue(Q, W, qsl, wsl, q0, 0, t);            // prologue: slab 0
    for (int s = 0; s < FEAT / 32; ++s) {
        const int cur = s & 1;
        if (s + 1 < FEAT / 32) {
            s1_issue(Q, W, qsl + (cur ^ 1) * QSL_SZ, wsl + (cur ^ 1) * WSL_SZ,
                     q0, (s + 1) * 32, t);
            WAIT_ASYNC(18);                        // slab s done; s+1 in flight
        } else {
            WAIT_ASYNC(0);
        }
        __syncthreads();
        const float* q = qsl + cur * QSL_SZ;
        const float* w = wsl + cur * WSL_SZ;
        #pragma unroll
        for (int kq = 0; kq < 32; kq += 4) {
            v2f a = *(const v2f*)&q[(mt1 * 16 + lm) * QP_ST + kq + ka];
            #pragma unroll
            for (int j = 0; j < 16; ++j) {
                int n = (ng1 + j) * 16 + lm;
                v2f b;
                b.x = w[(kq + ka + 0) * W_ST + n];
                b.y = w[(kq + ka + 1) * W_ST + n];
                acc[j] = wmma_f32(a, b, acc[j]);
            }
        }
        __syncthreads();                           // safe to overwrite buf cur
    }
    // epilogue: +bias, park z_query block in LDS
    #pragma unroll
    for (int j = 0; j < 16; ++j) {
        int n = (ng1 + j) * 16 + lm;
        float bv = bvec[n];
        #pragma unroll
        for (int r = 0; r < 8; ++r) {
            int row = mt1 * 16 + rbase + r;
            zq[row * ZQ_ST + n] = acc[j][r] + bv;
        }
    }
    __syncthreads();

    // q2 per row (4 threads per row)
    {
        int row = t >> 2, seg = t & 3;
        float s = 0.f;
        for (int i = 0; i < 128; ++i) {
            float v = zq[row * ZQ_ST + seg * 128 + i];
            s += v * v;
        }
        red[t] = s;
    }
    __syncthreads();
    if (t < 64) q2l[t] = red[4 * t] + red[4 * t + 1] + red[4 * t + 2] + red[4 * t + 3];

    // ---------------- stage 2: dot = zq @ zproto^T, K=512 ------------------
    #pragma unroll
    for (int j = 0; j < 8; ++j) acc[j] = vzero;
    const int mt2 = wv & 3;
    const int ng2 = (wv >> 2) * 8;

    s2_issue(zproto, psl, 0, t);
    for (int s = 0; s < EMB / 32; ++s) {
        const int cur = s & 1;
        if (s + 1 < EMB / 32) {
            s2_issue(zproto, psl + (cur ^ 1) * PSL_SZ, (s + 1) * 32, t);
            WAIT_ASYNC(8);
        } else {
            WAIT_ASYNC(0);
        }
        __syncthreads();
        const float* p = psl + cur * PSL_SZ;
        const int k0l = s * 32;
        #pragma unroll
        for (int kq = 0; kq < 32; kq += 4) {
            v2f a = *(const v2f*)&zq[(mt2 * 16 + lm) * ZQ_ST + k0l + kq + ka];
            #pragma unroll
            for (int j = 0; j < 8; ++j) {
                int n = (ng2 + j) * 16 + lm;
                v2f b = *(const v2f*)&p[n * QP_ST + kq + ka];
                acc[j] = wmma_f32(a, b, acc[j]);
            }
        }
        __syncthreads();
    }
    // scores = -sqrt(max(q2 + p2 - 2*dot, 0)) -> scorebuf
    #pragma unroll
    for (int j = 0; j < 8; ++j) {
        int n = (ng2 + j) * 16 + lm;
        float pp = p2[n];
        #pragma unroll
        for (int r = 0; r < 8; ++r) {
            int row = mt2 * 16 + rbase + r;
            float d2 = q2l[row] + pp - 2.f * acc[j][r];
            d2 = d2 > 0.f ? d2 : 0.f;
            sb[row * SB_ST + n] = -sqrtf(d2);
        }
    }
    __syncthreads();

    // ---------------- softmax over 256 ways per row ------------------------
    {
        int row = t >> 2, seg = t & 3;
        float m = -3.4e38f;
        for (int i = 0; i < 64; ++i) m = fmaxf(m, sb[row * SB_ST + seg * 64 + i]);
        red[t] = m;
    }
    __syncthreads();
    if (t < 64)
        rwm[t] = fmaxf(fmaxf(red[4 * t], red[4 * t + 1]),
                       fmaxf(red[4 * t + 2], red[4 * t + 3]));
    __syncthreads();
    {
        int row = t >> 2, seg = t & 3;
        float rm = rwm[row], s = 0.f;
        for (int i = 0; i < 64; ++i) s += __expf(sb[row * SB_ST + seg * 64 + i] - rm);
        red[t] = s;
    }
    __syncthreads();
    if (t < 64) rws[t] = red[4 * t] + red[4 * t + 1] + red[4 * t + 2] + red[4 * t + 3];
    __syncthreads();
    for (int m = 0; m < 64; ++m) {
        float v = __expf(sb[m * SB_ST + t] - rwm[m]) / rws[m];
        out[(size_t)(q0 + m) * WAYS + t] = v;      // coalesced 1KB stores
    }
}

// ---------------------------------------------------------------------------
extern "C" void kernel_launch(void* const* d_in, const int* in_sizes, int n_in,
                              void* d_out, int out_size, void* d_ws, size_t ws_size,
                              hipStream_t stream) {
    const float* S      = (const float*)d_in[0];
    const int*   onehot = (const int*)  d_in[1];
    const float* Q      = (const float*)d_in[2];
    const float* W      = (const float*)d_in[3];
    const float* bvec   = (const float*)d_in[4];
    float* out = (float*)d_out;

    char* ws = (char*)d_ws;
    int*   counts   = (int*)  (ws + 0);
    int*   labels   = (int*)  (ws + 1024);
    float* partials = (float*)(ws + 263168);
    float* smean    = (float*)(ws + 263168 + (size_t)CHUNKS * WAYS * FEAT * 4);
    float* zproto   = (float*)((char*)smean + (size_t)WAYS * FEAT * 4);
    float* p2       = (float*)((char*)zproto + (size_t)WAYS * EMB * 4);

    (void)hipFuncSetAttribute((const void*)k_class_partial,
                              hipFuncAttributeMaxDynamicSharedMemorySize,
                              WAYS * 256 * 4);
    (void)hipFuncSetAttribute((const void*)k_query_fused,
                              hipFuncAttributeMaxDynamicSharedMemorySize,
                              BC_LDS_B);

    k_zero_counts<<<1, 256, 0, stream>>>(counts);
    k_labels<<<NS / 256, 256, 0, stream>>>(onehot, labels, counts);
    k_class_partial<<<dim3(CHUNKS, FEAT / 256), 256, WAYS * 256 * 4, stream>>>(
        S, labels, partials);
    k_reduce_mean<<<(WAYS * FEAT) / 256, 256, 0, stream>>>(partials, counts, smean);
    k_proto_gemm<<<dim3(EMB / 16, WAYS / 16), 32, 0, stream>>>(smean, W, bvec, zproto);
    k_p2<<<1, WAYS, 0, stream>>>(zproto, p2);
    k_query_fused<<<NQ / 64, 256, BC_LDS_B, stream>>>(Q, W, bvec, zproto, p2, out);
}